// ShuffledGroupWhitening_82652350644756
// MI455X (gfx1250) — compile-verified
//
#include <hip/hip_runtime.h>

// ---------------------------------------------------------------------------
// Shuffled Group Whitening on gfx1250 (MI455X), wave32, WMMA f32 16x16x4.
//   x:     [3*8192, 1024] f32     perms: [3, 1024] i32
//   out:   [3*8192, 1024] f32
// d = 1024/64 = 16  ==> every matrix op is a native 16x16 WMMA tile.
//
// Memory-bound problem (FLOPs ~1.6G, bytes ~300MB); pipeline is 3 passes:
//   pass 1: raw second moment (WMMA XtX) + column sums   [read 100MB]
//   small : cov = M2 - mu mu^T; W = cov^{-1/2} (Newton-Schulz, WMMA in LDS)
//   pass 2: y = (x - mu) W, scatter through perm          [read+write 200MB]
// Whole working set (100MB) fits in the 192MB L2, so the permutation-gather
// access pattern amortizes across the 16 groups sharing each cacheline.
// ---------------------------------------------------------------------------

#define S_POS 3
#define BATCH 8192
#define D_FEAT 1024
#define G_GROUPS 64
#define GD 16
#define LDP 17           // LDS row pitch (pad to dodge bank conflicts)
#define NS_ITERS 8
#define COV_CHUNKS 32    // 256 rows per covariance wave
#define APPLY_CHUNKS 64  // 128 rows per apply wave

typedef __attribute__((ext_vector_type(2))) float v2f;
typedef __attribute__((ext_vector_type(8))) float v8f;

// ws layout (floats): [0,3072) column means ; [4096, 4096+192*256) cov->W
#define WS_MEAN_OFF 0
#define WS_COV_OFF 4096
#define WS_ZERO_FLOATS (WS_COV_OFF + S_POS * G_GROUPS * 256)

// ---------------------------------------------------------------------------
__global__ void zero_ws_kernel(float* __restrict__ ws) {
  int i = blockIdx.x * blockDim.x + threadIdx.x;
  if (i < WS_ZERO_FLOATS) ws[i] = 0.0f;
}

// ---------------------------------------------------------------------------
// Fused pass 1: raw M2[s,g] = (1/B) * X^T X via V_WMMA_F32_16X16X4_F32, plus
// per-column sums (means) from the same registers -- saves a whole 100MB pass.
// A-matrix 16x4 f32 layout: lanes 0-15 -> M=lane, VGPR0/1 = K0/K1;
//                           lanes 16-31 -> M=lane-16, VGPR0/1 = K2/K3.
// For X^T X the A and B fragments are the *same* register pair per lane.
__global__ void __launch_bounds__(32)
cov_kernel(const float* __restrict__ x, const int* __restrict__ perms,
           float* __restrict__ means, float* __restrict__ cov) {
  const int lane = threadIdx.x;
  const int s = blockIdx.z, g = blockIdx.y;
  const int m = lane & 15;
  const int kb = (lane < 16) ? 0 : 2;
  const int c = perms[s * D_FEAT + g * GD + m];   // gather column for feature m
  const int b0 = blockIdx.x * (BATCH / COV_CHUNKS);
  size_t base = ((size_t)s * BATCH + b0 + kb) * D_FEAT + c;
  v8f acc = {};
  float csum = 0.0f;
  for (int it = 0; it < (BATCH / COV_CHUNKS) / 4; ++it) {
    v2f a;
    a.x = x[base];
    a.y = x[base + D_FEAT];
    csum += a.x + a.y;
    acc = __builtin_amdgcn_wmma_f32_16x16x4_f32(false, a, false, a, (short)0,
                                                acc, false, false);
    base += (size_t)4 * D_FEAT;
  }
  // lanes L and L+16 each hold half the rows of column c -> both atomically add
  atomicAdd(&means[s * D_FEAT + c], csum * (1.0f / BATCH));
  float* cg = cov + (s * G_GROUPS + g) * 256;
  const int mrow = (lane < 16) ? 0 : 8;
#pragma unroll
  for (int v = 0; v < 8; ++v)
    atomicAdd(&cg[(v + mrow) * GD + m], acc[v] * (1.0f / BATCH));
}

// ---------------------------------------------------------------------------
// 16x16x16 f32 matmul of two row-major LDS matrices (pitch LDP), one wave.
__device__ __forceinline__ v8f wmma_mm16(const float* A, const float* B,
                                         int lane) {
  const int m = lane & 15;
  const int kb = (lane < 16) ? 0 : 2;
  v8f c = {};
#pragma unroll
  for (int k0 = 0; k0 < GD; k0 += 4) {
    v2f a, b;
    a.x = A[m * LDP + (k0 + kb)];
    a.y = A[m * LDP + (k0 + kb + 1)];
    b.x = B[(k0 + kb) * LDP + m];
    b.y = B[(k0 + kb + 1) * LDP + m];
    c = __builtin_amdgcn_wmma_f32_16x16x4_f32(false, a, false, b, (short)0, c,
                                              false, false);
  }
  return c;
}

// cov = M2 - mu mu^T ; W = cov^{-1/2} via trace-normalized Newton-Schulz:
//   Y0 = cov/tr, Z0 = I;  T = 3I - Z Y;  Y <- 0.5 Y T;  Z <- 0.5 T Z
//   cov^{-1/2} = Z / sqrt(tr)
__global__ void __launch_bounds__(32)
ns_invsqrt_kernel(const int* __restrict__ perms, const float* __restrict__ means,
                  float* __restrict__ covw) {
  __shared__ float Y[GD][LDP], Z[GD][LDP], T[GD][LDP];
  __shared__ float smu[GD];
  const int lane = threadIdx.x;
  const int n = lane & 15;
  const int m0 = (lane < 16) ? 0 : 8;
  const int sg = blockIdx.x;
  const int s = sg / G_GROUPS, g = sg % G_GROUPS;
  float* cg = covw + sg * 256;
  if (lane < GD)
    smu[lane] = means[s * D_FEAT + perms[s * D_FEAT + g * GD + lane]];
  __syncthreads();
#pragma unroll
  for (int v = 0; v < 8; ++v) {
    const int mm = v + m0;
    Y[mm][n] = cg[mm * GD + n] - smu[mm] * smu[n];   // center: M2 - mu mu^T
  }
  __syncthreads();
  float tr = 0.0f;
#pragma unroll
  for (int i = 0; i < GD; ++i) tr += Y[i][i];
  tr = fmaxf(tr, 1e-20f);
  const float itr = 1.0f / tr;
#pragma unroll
  for (int v = 0; v < 8; ++v) {
    const int mm = v + m0;
    Y[mm][n] *= itr;
    Z[mm][n] = (mm == n) ? 1.0f : 0.0f;
  }
  __syncthreads();
  for (int it = 0; it < NS_ITERS; ++it) {
    v8f zy = wmma_mm16(&Z[0][0], &Y[0][0], lane);
#pragma unroll
    for (int v = 0; v < 8; ++v) {
      const int mm = v + m0;
      T[mm][n] = ((mm == n) ? 3.0f : 0.0f) - zy[v];
    }
    __syncthreads();
    v8f yt = wmma_mm16(&Y[0][0], &T[0][0], lane);
    v8f tz = wmma_mm16(&T[0][0], &Z[0][0], lane);
    __syncthreads();
#pragma unroll
    for (int v = 0; v < 8; ++v) {
      const int mm = v + m0;
      Y[mm][n] = 0.5f * yt[v];
      Z[mm][n] = 0.5f * tz[v];
    }
    __syncthreads();
  }
  const float sc = rsqrtf(tr);
#pragma unroll
  for (int v = 0; v < 8; ++v) cg[(v + m0) * GD + n] = Z[v + m0][n] * sc;
}

// ---------------------------------------------------------------------------
// Pass 2: y = (x - mu) * W per 16-row batch tile; scatter-store to
// out[:, perm[g*16+n]]. W lives in B-fragments for the whole wave.
__global__ void __launch_bounds__(32)
apply_kernel(const float* __restrict__ x, const int* __restrict__ perms,
             const float* __restrict__ means, const float* __restrict__ W,
             float* __restrict__ out) {
  __shared__ int scol[GD];
  __shared__ float smu[GD];
  const int lane = threadIdx.x;
  const int s = blockIdx.z, g = blockIdx.y;
  const int n = lane & 15;
  const int kb = (lane < 16) ? 0 : 2;
  if (lane < GD) {
    const int c = perms[s * D_FEAT + g * GD + lane];
    scol[lane] = c;
    smu[lane] = means[s * D_FEAT + c];
  }
  __syncthreads();
  const float* Wg = W + (s * G_GROUPS + g) * 256;
  v2f bf[4];
#pragma unroll
  for (int j = 0; j < 4; ++j) {   // B fragment: B[K][N] = W[f][n]
    bf[j].x = Wg[(4 * j + kb) * GD + n];
    bf[j].y = Wg[(4 * j + kb + 1) * GD + n];
  }
  const int b0base = blockIdx.x * (BATCH / APPLY_CHUNKS);
  for (int t = 0; t < (BATCH / APPLY_CHUNKS) / GD; ++t) {
    const int b0 = b0base + t * GD;
    // A fragment: A[M][K] = Xc[b0+M][feat K];  M = lane&15
    const size_t rowbase = ((size_t)s * BATCH + b0 + n) * D_FEAT;
    v8f c = {};
#pragma unroll
    for (int j = 0; j < 4; ++j) {
      const int k0 = 4 * j + kb;
      v2f a;
      a.x = x[rowbase + scol[k0]] - smu[k0];
      a.y = x[rowbase + scol[k0 + 1]] - smu[k0 + 1];
      c = __builtin_amdgcn_wmma_f32_16x16x4_f32(false, a, false, bf[j],
                                                (short)0, c, false, false);
    }
    const int mrow = (lane < 16) ? 0 : 8;
    const size_t outbase = ((size_t)s * BATCH + b0) * D_FEAT + scol[n];
#pragma unroll
    for (int v = 0; v < 8; ++v)
      out[outbase + (size_t)(v + mrow) * D_FEAT] = c[v];
  }
}

// ---------------------------------------------------------------------------
extern "C" void kernel_launch(void* const* d_in, const int* in_sizes, int n_in,
                              void* d_out, int out_size, void* d_ws,
                              size_t ws_size, hipStream_t stream) {
  const float* x = (const float*)d_in[0];
  const int* perms = (const int*)d_in[1];
  float* out = (float*)d_out;
  float* ws = (float*)d_ws;
  float* means = ws + WS_MEAN_OFF;
  float* covw = ws + WS_COV_OFF;

  zero_ws_kernel<<<(WS_ZERO_FLOATS + 255) / 256, 256, 0, stream>>>(ws);
  cov_kernel<<<dim3(COV_CHUNKS, G_GROUPS, S_POS), 32, 0, stream>>>(x, perms,
                                                                   means, covw);
  ns_invsqrt_kernel<<<S_POS * G_GROUPS, 32, 0, stream>>>(perms, means, covw);
  apply_kernel<<<dim3(APPLY_CHUNKS, G_GROUPS, S_POS), 32, 0, stream>>>(
      x, perms, means, covw, out);
}